// RBFBias_17781164605640
// MI455X (gfx1250) — compile-verified
//
#include <hip/hip_runtime.h>

typedef __attribute__((ext_vector_type(2))) float v2f;
typedef __attribute__((ext_vector_type(8))) float v8f;

#define B_    2
#define H_    8
#define SQ_   2048
#define SK_   2048
#define QTILES (SQ_/16)
#define KTILES (SK_/16)
#define NLOG2E (-1.4426950408889634f)

__global__ __launch_bounds__(256) void rbf_bias_kernel(
    const float* __restrict__ score,
    const float* __restrict__ qs_s,
    const float* __restrict__ ks_s,
    const float* __restrict__ qs_t,
    const float* __restrict__ ks_t,
    const float* __restrict__ s_alpha,
    const float* __restrict__ s_beta,
    const float* __restrict__ t_alpha,
    const float* __restrict__ t_beta,
    float* __restrict__ out)
{
    const int lane = threadIdx.x & 31;
    const int wave = threadIdx.x >> 5;
    const int tile = blockIdx.x * 8 + wave;          // 32768 tiles total

    const int b    = tile >> 14;                     // / (128*128)
    const int rem  = tile & (QTILES * KTILES - 1);
    const int qt   = rem >> 7;                       // / 128
    const int kt   = rem & (KTILES - 1);
    const int qbase = qt * 16;
    const int kbase = kt * 16;

    const int n  = lane & 15;                        // k column within tile
    const int hi = lane >> 4;                        // half-wave select

    // ---- Load the 4 packed features [s0,s1,s2,t] for q-item and k-item (lane%16) ----
    const float* qsp = qs_s + ((size_t)b * SQ_ + qbase + n) * 3;
    const float  q0 = qsp[0], q1 = qsp[1], q2 = qsp[2];
    const float  qtv = qs_t[(size_t)b * SQ_ + qbase + n];

    const float* ksp = ks_s + ((size_t)b * SK_ + kbase + n) * 3;
    const float  k0 = ksp[0], k1 = ksp[1], k2 = ksp[2];
    const float  ktv = ks_t[(size_t)b * SK_ + kbase + n];

    const float qnorm_self = q0*q0 + q1*q1 + q2*q2;  // |q_n|^2 (s-features)
    const float knorm_self = k0*k0 + k1*k1 + k2*k2;  // |k_n|^2 (s-features)

    // ---- WMMA operand packing (16x4 f32 A layout: lanes 0-15 feats {0,1}, 16-31 feats {2,3}) ----
    v2f a, bb;
    a.x  = hi ? q2  : q0;
    a.y  = hi ? qtv : q1;
    bb.x = hi ? k2  : k0;
    bb.y = hi ? ktv : k1;

    v8f c = {};
    // c[r] = q_m . k_n over [s0,s1,s2,t]  (m = r + 8*hi, n = lane%16)
    c = __builtin_amdgcn_wmma_f32_16x16x4_f32(false, a, false, bb, (short)0, c,
                                              false, false);

    // ---- Per-row squared distances, shared across all 8 heads ----
    float ds2[8], dt2[8];
    const int idx_base = hi ? 32 : 0;                // byte index of source lane m = r + 8*hi
#pragma unroll
    for (int r = 0; r < 8; ++r) {
        const int idx = idx_base + r * 4;
        const float qn_m = __int_as_float(
            __builtin_amdgcn_ds_bpermute(idx, __float_as_int(qnorm_self)));
        const float qt_m = __int_as_float(
            __builtin_amdgcn_ds_bpermute(idx, __float_as_int(qtv)));
        const float sdot = c[r] - qt_m * ktv;        // remove t cross-term
        ds2[r] = qn_m + knorm_self - 2.0f * sdot;
        const float dt = qt_m - ktv;
        dt2[r] = dt * dt;
    }

    // ---- Stream score -> out for all 8 heads, 8 rows each ----
    const size_t bofs = (size_t)b * H_ * SQ_ * SK_;
    const int row0 = (qbase + hi * 8) * SK_ + kbase + n;   // element offset at r=0
    const float* sbase = score + bofs + row0;
    float*       obase = out   + bofs + row0;

#pragma unroll 1
    for (int h = 0; h < H_; ++h) {
        // uniform coefficients -> scalar loads
        const float sa0 = s_alpha[0*H_+h], sa1 = s_alpha[1*H_+h];
        const float sa2 = s_alpha[2*H_+h], sa3 = s_alpha[3*H_+h];
        const float nb0 = NLOG2E * s_beta[0*H_+h], nb1 = NLOG2E * s_beta[1*H_+h];
        const float nb2 = NLOG2E * s_beta[2*H_+h], nb3 = NLOG2E * s_beta[3*H_+h];
        const float ta0 = t_alpha[0*H_+h], ta1 = t_alpha[1*H_+h];
        const float ta2 = t_alpha[2*H_+h], ta3 = t_alpha[3*H_+h];
        const float tb0 = NLOG2E * t_beta[0*H_+h], tb1 = NLOG2E * t_beta[1*H_+h];
        const float tb2 = NLOG2E * t_beta[2*H_+h], tb3 = NLOG2E * t_beta[3*H_+h];

        const float* sp = sbase + (size_t)h * SQ_ * SK_;
        float*       op = obase + (size_t)h * SQ_ * SK_;

#pragma unroll
        for (int r = 0; r < 8; ++r) {
            float acc = sp[r * SK_];                 // r*SK*4 bytes -> 24-bit inst offset
            const float d = ds2[r];
            const float e = dt2[r];
            acc += sa0 * __builtin_amdgcn_exp2f(nb0 * d);
            acc += sa1 * __builtin_amdgcn_exp2f(nb1 * d);
            acc += sa2 * __builtin_amdgcn_exp2f(nb2 * d);
            acc += sa3 * __builtin_amdgcn_exp2f(nb3 * d);
            acc += ta0 * __builtin_amdgcn_exp2f(tb0 * e);
            acc += ta1 * __builtin_amdgcn_exp2f(tb1 * e);
            acc += ta2 * __builtin_amdgcn_exp2f(tb2 * e);
            acc += ta3 * __builtin_amdgcn_exp2f(tb3 * e);
            op[r * SK_] = acc;
        }
    }
}

extern "C" void kernel_launch(void* const* d_in, const int* in_sizes, int n_in,
                              void* d_out, int out_size, void* d_ws, size_t ws_size,
                              hipStream_t stream) {
    (void)in_sizes; (void)n_in; (void)out_size; (void)d_ws; (void)ws_size;
    const float* score   = (const float*)d_in[0];
    const float* qs_s    = (const float*)d_in[1];
    const float* ks_s    = (const float*)d_in[2];
    const float* qs_t    = (const float*)d_in[3];
    const float* ks_t    = (const float*)d_in[4];
    const float* s_alpha = (const float*)d_in[5];
    const float* s_beta  = (const float*)d_in[6];
    const float* t_alpha = (const float*)d_in[7];
    const float* t_beta  = (const float*)d_in[8];
    float* outp = (float*)d_out;

    // 32768 tiles of 16x16, 8 waves (tiles) per 256-thread block
    dim3 grid(4096), block(256);
    rbf_bias_kernel<<<grid, block, 0, stream>>>(score, qs_s, ks_s, qs_t, ks_t,
                                                s_alpha, s_beta, t_alpha, t_beta,
                                                outp);
}